// Quantizer_28235115004470
// MI455X (gfx1250) — compile-verified
//
#include <hip/hip_runtime.h>
#include <cstdint>

// ---------------------------------------------------------------------------
// K-quant group fitting for MI455X (gfx1250, wave32).
//  - Block = 256 threads handles 256 groups (= 32 rows, 32 KB of x).
//  - Tile staged via 8x GLOBAL_LOAD_ASYNC_TO_LDS_B128 per lane-set (ASYNCcnt),
//    one uniform bounds branch; addresses strength-reduce to adds.
//  - LDS group-major, stride 36 floats (144 B, 16B-aligned) -> ds_load_b128
//    reads at the 2-bank-cycle minimum (all 64 banks hit exactly twice).
//  - One lane owns one group of 32: 22-candidate search entirely in VGPRs,
//    4-way split accumulators for VOPD dual-issue, v_med3 clamp.
//  - Per-row (8-group) max reduction via wave32 __shfl_xor over width 8.
// ---------------------------------------------------------------------------

#define GROUP        32
#define MAXQ_F       15.0f
#define SMAXQ_F      63.0f
#define KQ_EPS       1e-9f
#define TPB          256
#define LDS_STRIDE   36            // floats; 144 B: 16B-aligned, good bank spread

__device__ __forceinline__ void async_ld_b32(unsigned ldsoff, uint64_t ga)
{
    asm volatile("global_load_async_to_lds_b32 %0, %1, off"
                 :: "v"(ldsoff), "v"(ga) : "memory");
}

__device__ __forceinline__ void async_ld_b128(unsigned ldsoff, uint64_t ga)
{
    asm volatile("global_load_async_to_lds_b128 %0, %1, off"
                 :: "v"(ldsoff), "v"(ga) : "memory");
}

// Weighted sums over quantized values: L = sum w*q, L2 = sum w*q*q, XL = sum w*q*x
// with q = med3(rint(x*a + b), 0, 15). 4-way split accumulators for ILP.
__device__ __forceinline__ void wsum3(const float* __restrict__ xv,
                                      const float* __restrict__ w,
                                      float a, float b,
                                      float& L, float& L2, float& XL)
{
    float l0 = 0.f, l1 = 0.f, l2_ = 0.f, l3 = 0.f;
    float q0 = 0.f, q1 = 0.f, q2 = 0.f, q3 = 0.f;
    float x0 = 0.f, x1 = 0.f, x2 = 0.f, x3 = 0.f;
    #pragma unroll
    for (int j = 0; j < GROUP; j += 4) {
        float t0 = __builtin_amdgcn_fmed3f(rintf(fmaf(xv[j+0], a, b)), 0.f, MAXQ_F);
        float t1 = __builtin_amdgcn_fmed3f(rintf(fmaf(xv[j+1], a, b)), 0.f, MAXQ_F);
        float t2 = __builtin_amdgcn_fmed3f(rintf(fmaf(xv[j+2], a, b)), 0.f, MAXQ_F);
        float t3 = __builtin_amdgcn_fmed3f(rintf(fmaf(xv[j+3], a, b)), 0.f, MAXQ_F);
        float w0 = w[j+0] * t0, w1 = w[j+1] * t1, w2 = w[j+2] * t2, w3 = w[j+3] * t3;
        l0 += w0;                     l1 += w1;
        l2_ += w2;                    l3 += w3;
        q0 = fmaf(w0, t0, q0);        q1 = fmaf(w1, t1, q1);
        q2 = fmaf(w2, t2, q2);        q3 = fmaf(w3, t3, q3);
        x0 = fmaf(w0, xv[j+0], x0);   x1 = fmaf(w1, xv[j+1], x1);
        x2 = fmaf(w2, xv[j+2], x2);   x3 = fmaf(w3, xv[j+3], x3);
    }
    L  = (l0 + l1) + (l2_ + l3);
    L2 = (q0 + q1) + (q2 + q3);
    XL = (x0 + x1) + (x2 + x3);
}

__global__ __launch_bounds__(TPB)
void kquant_kernel(const float* __restrict__ x, float* __restrict__ out,
                   int ngroups, long long n_elems)
{
    __shared__ float lds[TPB * LDS_STRIDE];

    const int tid  = threadIdx.x;
    const int gid  = blockIdx.x * TPB + tid;
    const long long base = (long long)blockIdx.x * (TPB * GROUP);

    // ---- async global -> LDS staging (gfx1250 ASYNCcnt path) ----
    // Fast path: whole 32 KB tile in range (block-uniform scalar test) ->
    // 8x GLOBAL_LOAD_ASYNC_TO_LDS_B128 per lane (512 B / wave / issue).
    if (base + (long long)(TPB * GROUP) <= n_elems) {
        const float4* x4 = (const float4*)(x + base);   // base % 8192 == 0 -> 16B aligned
        const int j4  = tid & 7;                        // float4 within group
        const int gq0 = tid >> 3;                       // base group for this lane
        #pragma unroll
        for (int k = 0; k < 8; ++k) {
            const int g = gq0 + 32 * k;                 // f = tid + k*256 -> g = f>>3
            unsigned ldsoff = (unsigned)(uintptr_t)(&lds[g * LDS_STRIDE + 4 * j4]);
            uint64_t ga     = (uint64_t)(uintptr_t)(x4 + tid + k * TPB);
            async_ld_b128(ldsoff, ga);
        }
    } else {
        // Tail path (cold): element-granular guarded b32 issues.
        const int g0 = tid >> 5;
        const int j  = tid & 31;
        #pragma unroll
        for (int k = 0; k < GROUP; ++k) {
            const long long ge = base + tid + k * TPB;
            if (ge < n_elems) {
                const int g = g0 + 8 * k;
                unsigned ldsoff = (unsigned)(uintptr_t)(&lds[g * LDS_STRIDE + j]);
                uint64_t ga     = (uint64_t)(uintptr_t)(x + ge);
                async_ld_b32(ldsoff, ga);
            }
        }
    }
    asm volatile("s_wait_asynccnt 0x0" ::: "memory");
    __syncthreads();

    if (gid >= ngroups) return;

    // ---- pull this lane's group as 8x ds_load_b128 (16B-aligned, stride 144B) ----
    float xv[GROUP];
    {
        const float4* gp = (const float4*)(&lds[tid * LDS_STRIDE]);
        #pragma unroll
        for (int q = 0; q < GROUP / 4; ++q) {
            const float4 v = gp[q];
            xv[4*q+0] = v.x; xv[4*q+1] = v.y; xv[4*q+2] = v.z; xv[4*q+3] = v.w;
        }
    }

    // ---- static per-group quantities ----
    float sum_x2 = 0.0f, xmx = xv[0], xmn = xv[0];
    #pragma unroll
    for (int j = 0; j < GROUP; ++j) {
        sum_x2 = fmaf(xv[j], xv[j], sum_x2);
        xmx = fmaxf(xmx, xv[j]);
        xmn = fminf(xmn, xv[j]);
    }
    const float xmin = fminf(xmn, 0.0f);
    const float av   = sqrtf(sum_x2 * (1.0f / GROUP));

    float w[GROUP];
    float sum_w = 0.0f, sum_x = 0.0f, sum_wx2 = 0.0f;
    #pragma unroll
    for (int j = 0; j < GROUP; ++j) {
        w[j] = av + fabsf(xv[j]);
        sum_w += w[j];
        const float wx = w[j] * xv[j];
        sum_x   += wx;                       // reference sum_x = sum(w*x)
        sum_wx2  = fmaf(wx, xv[j], sum_wx2);
    }

    const bool cflag = (xmx == xmin);
    const float scale  = cflag ? 0.0f : (xmx - xmin) * (1.0f / MAXQ_F);
    const float iscale = 1.0f / fmaxf(scale, KQ_EPS);

    // ---- initial fit ----
    float l, l2, xl;
    wsum3(xv, w, iscale, -xmin * iscale, l, l2, xl);
    if (cflag) { l = 0.0f; l2 = 0.0f; xl = 0.0f; }

    float best_scale = scale, best_min = xmin;
    // err = s^2*l2 + m^2*sum_w + sum_wx2 + 2*(s*m*l - s*xl - m*sum_x)
    float best_err = scale * scale * l2 + xmin * xmin * sum_w + sum_wx2
                   + 2.0f * (scale * xmin * l - scale * xl - xmin * sum_x);

    const float rng = fmaxf(xmx - xmin, KQ_EPS);

    // ---- candidate search (rolled: 21 iters of ~190 unrolled VALU) ----
    #pragma unroll 1
    for (int i = 0; i <= 20; ++i) {
        // (RMIN + RDELTA*i + MAXQ) evaluated in double like the Python scalar path
        const float ci = (float)(14.0 + 0.1 * (double)i) / rng;
        float L, L2, XL;
        wsum3(xv, w, ci, -xmin * ci, L, L2, XL);
        if (cflag) { L = 0.0f; L2 = 0.0f; XL = 0.0f; }

        const float D     = sum_w * L2 - L * L;
        const bool  valid = (D > KQ_EPS);
        const float Ds    = valid ? D : 1.0f;
        float ts = (sum_w * XL - sum_x * L) / Ds;
        float tm = (L2 * sum_x - L * XL) / Ds;
        if (tm > 0.0f) { ts = XL / fmaxf(L2, KQ_EPS); tm = 0.0f; }

        const float cerr = ts * ts * L2 + tm * tm * sum_w + sum_wx2
                         + 2.0f * (ts * tm * L - ts * XL - tm * sum_x);
        if (valid && (cerr < best_err)) {
            best_err   = cerr;
            best_scale = ts;
            best_min   = tm;
        }
    }

    const float gscale = best_scale;
    const float gzero  = -best_min;

    // ---- per-row (8 groups) max via wave32 xor-shuffles, width 8 ----
    float ms = gscale, mz = gzero;
    #pragma unroll
    for (int off = 4; off >= 1; off >>= 1) {
        ms = fmaxf(ms, __shfl_xor(ms, off, 8));
        mz = fmaxf(mz, __shfl_xor(mz, off, 8));
    }

    const float sg_s = (float)(_Float16)(ms * (1.0f / SMAXQ_F));   // fp16 RNE round-trip
    const float sg_z = (float)(_Float16)(mz * (1.0f / SMAXQ_F));
    const float inv_s = (ms > 0.0f) ? SMAXQ_F / fmaxf(ms, KQ_EPS) : 0.0f;
    const float inv_z = (mz > 0.0f) ? SMAXQ_F / fmaxf(mz, KQ_EPS) : 0.0f;
    const float gsq = __builtin_amdgcn_fmed3f(rintf(inv_s * gscale), 0.0f, SMAXQ_F);
    const float gzq = __builtin_amdgcn_fmed3f(rintf(inv_z * gzero),  0.0f, SMAXQ_F);

    // ---- output: per row, 18 floats: [sg_s, gsq[8], sg_z, gzq[8]] ----
    const long long row = (long long)(gid >> 3);
    const int k = gid & 7;
    float* orow = out + row * 18;
    orow[1 + k]  = gsq;
    orow[10 + k] = gzq;
    if (k == 0) {
        orow[0] = sg_s;
        orow[9] = sg_z;
    }
}

extern "C" void kernel_launch(void* const* d_in, const int* in_sizes, int n_in,
                              void* d_out, int out_size, void* d_ws, size_t ws_size,
                              hipStream_t stream)
{
    const float* x  = (const float*)d_in[0];   // (rows, 256) fp32
    float* out      = (float*)d_out;           // (rows, 18)  fp32
    const long long n_elems = (long long)in_sizes[0];
    const int ngroups = (int)(n_elems / GROUP);
    const int blocks  = (ngroups + TPB - 1) / TPB;
    kquant_kernel<<<blocks, TPB, 0, stream>>>(x, out, ngroups, n_elems);
}